// ReceptiveFieldNorm_1580547966583
// MI455X (gfx1250) — compile-verified
//
#include <hip/hip_runtime.h>
#include <cmath>

typedef __attribute__((ext_vector_type(2))) float v2f;
typedef __attribute__((ext_vector_type(8))) float v8f;

#define IMG_H 768
#define IMG_W 768
#define SH    256
#define N_IMG 16
#define N_CH  3
#define SPIX  (SH * SH)           // 65536 coarse pixels per image
#define P_ELEMS (N_IMG * SPIX)    // 1,048,576 floats per coarse buffer
#define FPIX  (IMG_H * IMG_W)     // 589,824 fine pixels per plane

// ---- f32 WMMA 16x16x4 wrapper (CDNA5, wave32) ------------------------------
__device__ __forceinline__ v8f wmma4(v2f a, v2f b, v8f c) {
  // 8 args: (neg_a, A, neg_b, B, c_mod, C, reuse_a, reuse_b)
  return __builtin_amdgcn_wmma_f32_16x16x4_f32(false, a, false, b, (short)0, c,
                                               false, false);
}

// ---- Pass 1: strided "bilinear" 3x downsample + channel moments ------------
// xs[i,j] = x[3i+1, 3j+1]; u = sum_c xs, v = sum_c xs^2
__global__ void k_downsample(const float* x, float* __restrict__ u,
                             float* __restrict__ v) {
  int idx = blockIdx.x * blockDim.x + threadIdx.x;     // [0, P_ELEMS)
  int j = idx & 255;
  int i = (idx >> 8) & 255;
  int n = idx >> 16;
  int gy = 3 * i + 1, gx = 3 * j + 1;
  const float* base = x + (size_t)n * N_CH * FPIX + (size_t)gy * IMG_W + gx;
  float s = 0.f, ss = 0.f;
#pragma unroll
  for (int c = 0; c < N_CH; ++c) {
    float t = base[(size_t)c * FPIX];
    s += t;
    ss += t * t;
  }
  u[idx] = s;
  v[idx] = ss;
}

// ---- Horizontal box sum as banded matmul: OUT = X * W ----------------------
// W[k][n] = (|n-k| <= p).  One wave per 16x16 output tile; tile coords are
// forced scalar (readfirstlane) so the K-loop is wave-uniform (EXEC all 1s).
__global__ void k_hbox(const float* in0, const float* in1, float* out0,
                       float* out1, int p) {
  const float* in = blockIdx.y ? in1 : in0;
  float* out = blockIdx.y ? out1 : out0;
  int wv = __builtin_amdgcn_readfirstlane(threadIdx.x >> 5);   // scalar
  int wave = blockIdx.x * 4 + wv;                              // [0,4096)
  int img = wave >> 8;
  int tile = wave & 255;
  int m0 = (tile >> 4) << 4;
  int n0 = (tile & 15) << 4;
  int lane = threadIdx.x & 31;
  int m = lane & 15;          // A row / D col index
  int n = m;                  // B/D col index
  int half = lane >> 4;
  int kb = half << 1;         // lane-half K split {0,1} / {2,3}
  const float* X = in + (size_t)img * SPIX;
  const float* rowA = X + (m0 + m) * SH + kb;  // lane's A row, pre-offset

  int kLo = n0 - p;
  if (kLo < 0) kLo = 0;
  kLo &= ~3;                                   // scalar
  int kHi = n0 + 15 + p;
  if (kHi > SH - 1) kHi = SH - 1;              // scalar

  v8f acc = {0.f, 0.f, 0.f, 0.f, 0.f, 0.f, 0.f, 0.f};
  for (int k0 = kLo; k0 <= kHi; k0 += 4) {     // uniform scalar loop
    int ka = k0 + kb;                          // ka, ka+1 <= 255 (k0 <= 252)
    v2f a, b;
    a.x = rowA[k0];
    a.y = rowA[k0 + 1];
    int d0 = (n0 + n) - ka;
    b.x = (d0 <= p && d0 >= -p) ? 1.0f : 0.0f;
    int d1 = d0 - 1;
    b.y = (d1 <= p && d1 >= -p) ? 1.0f : 0.0f;
    acc = wmma4(a, b, acc);
  }
  float* O = out + (size_t)img * SPIX + (m0 + (half << 3)) * SH + (n0 + n);
#pragma unroll
  for (int r = 0; r < 8; ++r)                  // D: VGPR r -> M = r + 8*half
    O[r * SH] = acc[r];
}

// ---- Vertical box sum as banded matmul: OUT = W * X ------------------------
__global__ void k_vbox(const float* in0, const float* in1, float* out0,
                       float* out1, int p) {
  const float* in = blockIdx.y ? in1 : in0;
  float* out = blockIdx.y ? out1 : out0;
  int wv = __builtin_amdgcn_readfirstlane(threadIdx.x >> 5);   // scalar
  int wave = blockIdx.x * 4 + wv;
  int img = wave >> 8;
  int tile = wave & 255;
  int m0 = (tile >> 4) << 4;
  int n0 = (tile & 15) << 4;
  int lane = threadIdx.x & 31;
  int m = lane & 15;
  int n = m;
  int half = lane >> 4;
  int kb = half << 1;
  const float* X = in + (size_t)img * SPIX;
  const float* colB = X + kb * SH + (n0 + n);  // lane's B column, pre-offset

  int kLo = m0 - p;
  if (kLo < 0) kLo = 0;
  kLo &= ~3;
  int kHi = m0 + 15 + p;
  if (kHi > SH - 1) kHi = SH - 1;

  v8f acc = {0.f, 0.f, 0.f, 0.f, 0.f, 0.f, 0.f, 0.f};
  for (int k0 = kLo; k0 <= kHi; k0 += 4) {     // uniform scalar loop
    int ka = k0 + kb;
    v2f a, b;
    int d0 = (m0 + m) - ka;                    // A = banded W tile
    a.x = (d0 <= p && d0 >= -p) ? 1.0f : 0.0f;
    int d1 = d0 - 1;
    a.y = (d1 <= p && d1 >= -p) ? 1.0f : 0.0f;
    b.x = colB[k0 * SH];
    b.y = colB[k0 * SH + SH];
    acc = wmma4(a, b, acc);
  }
  float* O = out + (size_t)img * SPIX + (m0 + (half << 3)) * SH + (n0 + n);
#pragma unroll
  for (int r = 0; r < 8; ++r)
    O[r * SH] = acc[r];
}

// ---- Stats: U,V (2D box sums of u,v) -> a = 1/std, b = -mean/std -----------
__global__ void k_stats(const float* __restrict__ U, const float* __restrict__ V,
                        float* __restrict__ a, float* __restrict__ b, int p) {
  int idx = blockIdx.x * blockDim.x + threadIdx.x;
  int j = idx & 255;
  int i = (idx >> 8) & 255;
  int lo_i = i - p < 0 ? 0 : i - p, hi_i = i + p > SH - 1 ? SH - 1 : i + p;
  int lo_j = j - p < 0 ? 0 : j - p, hi_j = j + p > SH - 1 ? SH - 1 : j + p;
  float Minv = 1.0f / (float)((hi_i - lo_i + 1) * (hi_j - lo_j + 1));
  float inv3M = Minv * (1.0f / 3.0f);
  float xm = U[idx] * inv3M;
  float x2m = V[idx] * inv3M;
  float var = x2m - xm * xm;
  if (var < 0.f) var = 0.f;
  var += 1e-3f;
  float inv_std = 1.0f / sqrtf(var);
  a[idx] = inv_std;
  b[idx] = -xm * inv_std;
}

// ---- Divide box sums of a,b by window count M (in place) -------------------
__global__ void k_meandiv(float* A, float* B, int p) {
  int idx = blockIdx.x * blockDim.x + threadIdx.x;
  int j = idx & 255;
  int i = (idx >> 8) & 255;
  int lo_i = i - p < 0 ? 0 : i - p, hi_i = i + p > SH - 1 ? SH - 1 : i + p;
  int lo_j = j - p < 0 ? 0 : j - p, hi_j = j + p > SH - 1 ? SH - 1 : j + p;
  float Minv = 1.0f / (float)((hi_i - lo_i + 1) * (hi_j - lo_j + 1));
  A[idx] *= Minv;
  B[idx] *= Minv;
}

// ---- 3-phase bilinear upsample of (ma,mb) + affine apply -------------------
__global__ void k_apply(const float* xin, const float* __restrict__ Amap,
                        const float* __restrict__ Bmap, float* xout) {
  int idx = blockIdx.x * blockDim.x + threadIdx.x;  // [0, N_IMG*FPIX)
  int j = idx % IMG_W;
  int i = (idx / IMG_W) % IMG_H;
  int n = idx / FPIX;

  // src coord = (i-1)/3 ; fixed phase weights, edge-clamped taps
  int ay = i - 1;
  int iy = (ay >= 0) ? (ay / 3) : -1;
  float fy = (float)(ay - 3 * iy) * (1.0f / 3.0f);
  int y0 = iy < 0 ? 0 : iy;
  int y1 = iy + 1 > SH - 1 ? SH - 1 : iy + 1;
  int ax = j - 1;
  int ix = (ax >= 0) ? (ax / 3) : -1;
  float fx = (float)(ax - 3 * ix) * (1.0f / 3.0f);
  int x0 = ix < 0 ? 0 : ix;
  int x1 = ix + 1 > SH - 1 ? SH - 1 : ix + 1;

  size_t base = (size_t)n * N_CH * FPIX + (size_t)i * IMG_W + j;
  __builtin_prefetch(xin + base + 2 * (size_t)FPIX, 0, 0);  // global_prefetch_b8

  const float* Ai = Amap + (size_t)n * SPIX;
  const float* Bi = Bmap + (size_t)n * SPIX;
  float w00 = (1.f - fy) * (1.f - fx), w01 = (1.f - fy) * fx;
  float w10 = fy * (1.f - fx), w11 = fy * fx;
  float ma = w00 * Ai[y0 * SH + x0] + w01 * Ai[y0 * SH + x1] +
             w10 * Ai[y1 * SH + x0] + w11 * Ai[y1 * SH + x1];
  float mb = w00 * Bi[y0 * SH + x0] + w01 * Bi[y0 * SH + x1] +
             w10 * Bi[y1 * SH + x0] + w11 * Bi[y1 * SH + x1];
#pragma unroll
  for (int c = 0; c < N_CH; ++c) {
    size_t o = base + (size_t)c * FPIX;
    xout[o] = ma * xin[o] + mb;
  }
}

extern "C" void kernel_launch(void* const* d_in, const int* in_sizes, int n_in,
                              void* d_out, int out_size, void* d_ws,
                              size_t ws_size, hipStream_t stream) {
  const float* x = (const float*)d_in[0];
  float* out = (float*)d_out;
  float* w = (float*)d_ws;
  float* w0 = w;                           // u  -> a
  float* w1 = w + 1 * (size_t)P_ELEMS;     // v  -> b
  float* w2 = w + 2 * (size_t)P_ELEMS;     // hbox(u) -> hbox(a)
  float* w3 = w + 3 * (size_t)P_ELEMS;     // hbox(v) -> hbox(b)
  float* w4 = w + 4 * (size_t)P_ELEMS;     // U -> A
  float* w5 = w + 5 * (size_t)P_ELEMS;     // V -> B

  const int pvals[2] = {25, 6};  // iter1: win 51 (p=25); iter2: win 13 (p=6)
  dim3 bg((N_IMG * 256) / 4, 2); // 4096 tiles, 4 waves/block, 2 arrays

  for (int it = 0; it < 2; ++it) {
    int p = pvals[it];
    const float* src = (it == 0) ? x : (const float*)out;
    k_downsample<<<P_ELEMS / 256, 256, 0, stream>>>(src, w0, w1);
    k_hbox<<<bg, 128, 0, stream>>>(w0, w1, w2, w3, p);
    k_vbox<<<bg, 128, 0, stream>>>(w2, w3, w4, w5, p);
    k_stats<<<P_ELEMS / 256, 256, 0, stream>>>(w4, w5, w0, w1, p);
    k_hbox<<<bg, 128, 0, stream>>>(w0, w1, w2, w3, p);
    k_vbox<<<bg, 128, 0, stream>>>(w2, w3, w4, w5, p);
    k_meandiv<<<P_ELEMS / 256, 256, 0, stream>>>(w4, w5, p);
    k_apply<<<(N_IMG * FPIX) / 256, 256, 0, stream>>>(src, w4, w5, out);
  }
}